// real_KW4_52699248721959
// MI455X (gfx1250) — compile-verified
//
#include <hip/hip_runtime.h>

#ifndef __has_builtin
#define __has_builtin(x) 0
#endif

#define CLS   1000      // classes per row
#define BLOCK 256       // 8 wave32s on gfx1250
#define NWAVE (BLOCK / 32)

typedef int v4i __attribute__((ext_vector_type(4)));
typedef __attribute__((address_space(1))) v4i* gptr_b128;
typedef __attribute__((address_space(3))) v4i* lptr_b128;

static __device__ __forceinline__ void async_copy_b128(const float* g, float* l) {
#if __has_builtin(__builtin_amdgcn_global_load_async_to_lds_b128)
  // CDNA5 GLOBAL_LOAD_ASYNC_TO_LDS_B128: HBM -> LDS with no VGPR staging,
  // tracked by ASYNCcnt. Signature (from clang diagnostic): (v4i AS1*, v4i AS3*, imm, imm).
  __builtin_amdgcn_global_load_async_to_lds_b128((gptr_b128)g, (lptr_b128)l, 0, 0);
#else
  typedef __attribute__((ext_vector_type(4))) float f4;
  *(f4*)l = *(const f4*)g;
#endif
}

static __device__ __forceinline__ void wait_async_done() {
#if __has_builtin(__builtin_amdgcn_s_wait_asynccnt)
  __builtin_amdgcn_s_wait_asynccnt(0);
#elif __has_builtin(__builtin_amdgcn_global_load_async_to_lds_b128)
  asm volatile("s_wait_asynccnt 0" ::: "memory");
#endif
}

// Block-wide reduction of n values (n <= 5): wave32 shfl_xor tree, then
// cross-wave combine through a tiny LDS scratch. Results broadcast to all threads.
template <bool IS_MAX>
static __device__ __forceinline__ void block_reduce(float* v, int n, float* red,
                                                    int wave, int lane) {
  for (int i = 0; i < n; ++i) {
    float x = v[i];
    #pragma unroll
    for (int off = 16; off > 0; off >>= 1) {
      float y = __shfl_xor(x, off, 32);
      x = IS_MAX ? fmaxf(x, y) : (x + y);
    }
    v[i] = x;
  }
  __syncthreads();                       // prior readers of `red` are done
  if (lane == 0)
    for (int i = 0; i < n; ++i) red[i * NWAVE + wave] = v[i];
  __syncthreads();
  for (int i = 0; i < n; ++i) {
    float x = red[i * NWAVE];
    #pragma unroll
    for (int w = 1; w < NWAVE; ++w) {
      float y = red[i * NWAVE + w];
      x = IS_MAX ? fmaxf(x, y) : (x + y);
    }
    v[i] = x;
  }
}

__global__ __launch_bounds__(BLOCK) void kw_distill_kernel(
    const float* __restrict__ o1, const float* __restrict__ o2,
    const float* __restrict__ o3, const float* __restrict__ o4,
    float* __restrict__ out) {
  __shared__ float xs0[CLS], xs1[CLS], xs2[CLS], xs3[CLS];  // logits -> exp()
  __shared__ float emx[CLS];                                // mimic exp()
  __shared__ float red[5 * NWAVE];

  const int row  = blockIdx.x;
  const int t    = threadIdx.x;
  const int wave = t >> 5;
  const int lane = t & 31;
  const size_t base = (size_t)row * CLS;
  const float invT = 1.0f / 20.0f;

  // ---- Stage the row once from HBM into LDS (16 KB): 250 lanes x 16 B x 4 tensors
  if (t < CLS / 4) {
    const int c4 = t * 4;
    async_copy_b128(o1 + base + c4, xs0 + c4);
    async_copy_b128(o2 + base + c4, xs1 + c4);
    async_copy_b128(o3 + base + c4, xs2 + c4);
    async_copy_b128(o4 + base + c4, xs3 + c4);
  }
  wait_async_done();
  __syncthreads();

  // ---- Pass A: maxima of the 5 logit rows (max(x/T) == max(x)/T, fold into exp)
  float mx[5] = {-3.402823466e38f, -3.402823466e38f, -3.402823466e38f,
                 -3.402823466e38f, -3.402823466e38f};
  for (int c = t; c < CLS; c += BLOCK) {
    float a = xs0[c], b = xs1[c], d = xs2[c], e = xs3[c];
    float m = (a + b + d + e) * 0.25f;
    mx[0] = fmaxf(mx[0], a); mx[1] = fmaxf(mx[1], b);
    mx[2] = fmaxf(mx[2], d); mx[3] = fmaxf(mx[3], e);
    mx[4] = fmaxf(mx[4], m);
  }
  block_reduce<true>(mx, 5, red, wave, lane);

  // ---- Pass B: exp((x - max)/T), partition functions; cache exps in LDS
  float sm[5] = {0.f, 0.f, 0.f, 0.f, 0.f};
  for (int c = t; c < CLS; c += BLOCK) {
    float a = xs0[c], b = xs1[c], d = xs2[c], e = xs3[c];
    float m = (a + b + d + e) * 0.25f;
    float e0 = __expf((a - mx[0]) * invT);
    float e1 = __expf((b - mx[1]) * invT);
    float e2 = __expf((d - mx[2]) * invT);
    float e3 = __expf((e - mx[3]) * invT);
    float e4 = __expf((m - mx[4]) * invT);
    sm[0] += e0; sm[1] += e1; sm[2] += e2; sm[3] += e3; sm[4] += e4;
    xs0[c] = e0; xs1[c] = e1; xs2[c] = e2; xs3[c] = e3; emx[c] = e4;
  }
  block_reduce<false>(sm, 5, red, wave, lane);

  const float r0 = 1.0f / sm[0], r1 = 1.0f / sm[1], r2 = 1.0f / sm[2],
              r3 = 1.0f / sm[3], rm = 1.0f / sm[4];

  // ---- Pass C: sum of L1 distances to the mimic softmax
  float acc = 0.0f;
  for (int c = t; c < CLS; c += BLOCK) {
    float pm = emx[c] * rm;
    acc += fabsf(xs0[c] * r0 - pm) + fabsf(xs1[c] * r1 - pm) +
           fabsf(xs2[c] * r2 - pm) + fabsf(xs3[c] * r3 - pm);
  }
  block_reduce<false>(&acc, 1, red, wave, lane);

  if (t == 0) out[row] = acc * 0.25f;
}

extern "C" void kernel_launch(void* const* d_in, const int* in_sizes, int n_in,
                              void* d_out, int out_size, void* d_ws, size_t ws_size,
                              hipStream_t stream) {
  (void)d_ws; (void)ws_size; (void)out_size;
  const float* o1 = (const float*)d_in[0];
  const float* o2 = (const float*)d_in[1];
  const float* o3 = (const float*)d_in[2];
  const float* o4 = (const float*)d_in[3];
  float* out = (float*)d_out;

  const int B = in_sizes[0] / CLS;
  const int nprev = (n_in > 4) ? in_sizes[4] : 0;   // real_KW prefix (empty here)
  if (nprev > 0) {
    (void)hipMemcpyAsync(out, d_in[4], (size_t)nprev * sizeof(float),
                         hipMemcpyDeviceToDevice, stream);
  }
  kw_distill_kernel<<<B, BLOCK, 0, stream>>>(o1, o2, o3, o4, out + nprev);
}